// HPG_Mamba_Stage_76476187673192
// MI455X (gfx1250) — compile-verified
//
#include <hip/hip_runtime.h>
#include <math.h>

#define D_MODELC 96
#define D_STATEC 16
#define D_CONVC  4
#define D_INNERC 192
#define DT_RANKC 6
#define NBATCH   4
#define HWSZ     4096
#define LLEN     4096
#define NDBL     38      /* DT_RANK + 2*D_STATE */
#define EPSV     1e-5f

typedef float v2f __attribute__((ext_vector_type(2)));
typedef float v8f __attribute__((ext_vector_type(8)));

// D = A(16xK) * B(Kx16) + C via V_WMMA_F32_16X16X4_F32 steps.
__device__ __forceinline__ v8f wmma4(v2f a, v2f b, v8f c) {
  return __builtin_amdgcn_wmma_f32_16x16x4_f32(false, a, false, b, (short)0, c,
                                               false, false);
}

__device__ __forceinline__ float sigm_(float x) { return 1.f / (1.f + __expf(-x)); }
__device__ __forceinline__ float silu_(float x) { return x * sigm_(x); }

// cross-scan gather: (dir, l) -> hw index into (H,W)=64x64
__device__ __forceinline__ int hwmap(int dir, int l) {
  if (dir == 1 || dir == 3) l = LLEN - 1 - l;
  if (dir >= 2) return ((l & 63) << 6) | (l >> 6);
  return l;
}

// ---------- K1: 1x1 conv (GEMM) for both proj branches, bias fused ----------
__global__ void k_proj_in(const float* __restrict__ Fs, const float* __restrict__ HFs,
                          const float* __restrict__ pfw, const float* __restrict__ pfb,
                          const float* __restrict__ phw, const float* __restrict__ phb,
                          float* __restrict__ t1) {
  const int lane = threadIdx.x, lr = lane & 15, half = lane >> 4;
  const int p0 = blockIdx.x * 16;
  const int b = p0 >> 12, hw0 = p0 & 4095;
  const int o0 = blockIdx.y * 16;
  const int br = blockIdx.z;
  const float* X  = br ? HFs : Fs;
  const float* W  = br ? phw : pfw;
  const float* Bi = br ? phb : pfb;
  float* out = t1 + (size_t)br * NBATCH * D_MODELC * HWSZ;
  v8f acc;
  for (int r = 0; r < 8; ++r) acc[r] = Bi[o0 + r + 8 * half];
  const float* Wr = W + (size_t)(o0 + lr) * D_MODELC;
  const float* Xb = X + (size_t)b * D_MODELC * HWSZ + hw0 + lr;
  for (int k = 0; k < D_MODELC; k += 4) {
    v2f a, bb;
    a.x  = Wr[k + 2 * half];
    a.y  = Wr[k + 2 * half + 1];
    bb.x = Xb[(size_t)(k + 2 * half) * HWSZ];
    bb.y = Xb[(size_t)(k + 2 * half + 1) * HWSZ];
    acc = wmma4(a, bb, acc);
  }
  for (int r = 0; r < 8; ++r) {
    int o = o0 + r + 8 * half;
    out[((size_t)b * D_MODELC + o) * HWSZ + hw0 + lr] = acc[r];
  }
}

// ---------- K2: 3x3 depthwise conv + bias + SiLU (both branches) ----------
__global__ void k_dwconv(const float* __restrict__ t1,
                         const float* __restrict__ pfdw, const float* __restrict__ pfb2,
                         const float* __restrict__ phdw, const float* __restrict__ phb2,
                         float* __restrict__ Pf, float* __restrict__ Ph) {
  int idx = blockIdx.x * blockDim.x + threadIdx.x;
  int br = blockIdx.y;
  if (idx >= NBATCH * D_MODELC * HWSZ) return;
  int hw = idx & 4095, bc = idx >> 12, c = bc % D_MODELC;
  int h = hw >> 6, w = hw & 63;
  const float* in = t1 + (size_t)br * NBATCH * D_MODELC * HWSZ + (size_t)bc * HWSZ;
  const float* dw = (br ? phdw : pfdw) + c * 9;
  float s = (br ? phb2 : pfb2)[c];
  for (int ky = 0; ky < 3; ++ky) {
    int hy = h + ky - 1;
    if (hy < 0 || hy >= 64) continue;
    for (int kx = 0; kx < 3; ++kx) {
      int wx = w + kx - 1;
      if (wx < 0 || wx >= 64) continue;
      s += dw[ky * 3 + kx] * in[hy * 64 + wx];
    }
  }
  (br ? Ph : Pf)[idx] = silu_(s);
}

// ---------- K3: instance norm * G_s * gamma (in place on Ph) ----------
__global__ void k_instnorm(float* __restrict__ Ph, const float* __restrict__ Gs,
                           const float* __restrict__ gamma) {
  __shared__ float ssum[256], ssq[256];
  int bc = blockIdx.x;
  float* p = Ph + (size_t)bc * HWSZ;
  const float* g = Gs + (size_t)bc * HWSZ;
  float s = 0.f, sq = 0.f;
  for (int i = threadIdx.x; i < HWSZ; i += 256) { float v = p[i]; s += v; sq += v * v; }
  ssum[threadIdx.x] = s; ssq[threadIdx.x] = sq;
  __syncthreads();
  for (int off = 128; off > 0; off >>= 1) {
    if ((int)threadIdx.x < off) {
      ssum[threadIdx.x] += ssum[threadIdx.x + off];
      ssq[threadIdx.x]  += ssq[threadIdx.x + off];
    }
    __syncthreads();
  }
  float mu = ssum[0] / HWSZ;
  float var = ssq[0] / HWSZ - mu * mu;
  float rstd = rsqrtf(var + EPSV);
  float gam = gamma[0];
  for (int i = threadIdx.x; i < HWSZ; i += 256)
    p[i] = gam * (p[i] - mu) * rstd * g[i];
}

// ---------- K4: gate GEMM (96->96) + sigmoid + Xf multiply; cross-scan fused ----------
__global__ void k_gate(const float* __restrict__ Pf, const float* __restrict__ Phb,
                       const float* __restrict__ hfw, const float* __restrict__ hfb,
                       float* __restrict__ xm) {
  const int lane = threadIdx.x, lr = lane & 15, half = lane >> 4;
  const int l0 = blockIdx.x * 16, o0 = blockIdx.y * 16;
  const int dirb = blockIdx.z, dir = dirb >> 2, b = dirb & 3;
  const int hwn = hwmap(dir, l0 + lr);
  v8f acc;
  for (int r = 0; r < 8; ++r) acc[r] = hfb[dir * D_MODELC + o0 + r + 8 * half];
  const float* Wr = hfw + ((size_t)dir * D_MODELC + (o0 + lr)) * D_MODELC;
  const float* Xb = Phb + (size_t)b * D_MODELC * HWSZ + hwn;
  for (int k = 0; k < D_MODELC; k += 4) {
    v2f a, bb;
    a.x  = Wr[k + 2 * half];
    a.y  = Wr[k + 2 * half + 1];
    bb.x = Xb[(size_t)(k + 2 * half) * HWSZ];
    bb.y = Xb[(size_t)(k + 2 * half + 1) * HWSZ];
    acc = wmma4(a, bb, acc);
  }
  for (int r = 0; r < 8; ++r) {
    int o = o0 + r + 8 * half;
    float g  = sigm_(acc[r]);
    float xf = Pf[((size_t)b * D_MODELC + o) * HWSZ + hwn];
    xm[((size_t)dirb * D_MODELC + o) * LLEN + l0 + lr] = xf * g;
  }
}

// ---------- K5: in-projection GEMM (96 -> 384) ----------
__global__ void k_inproj(const float* __restrict__ xm, const float* __restrict__ inw,
                         float* __restrict__ xz) {
  const int lane = threadIdx.x, lr = lane & 15, half = lane >> 4;
  const int l0 = blockIdx.x * 16, o0 = blockIdx.y * 16;
  const int dirb = blockIdx.z, dir = dirb >> 2;
  v8f acc;
  for (int r = 0; r < 8; ++r) acc[r] = 0.f;
  const float* Wr = inw + ((size_t)dir * 2 * D_INNERC + (o0 + lr)) * D_MODELC;
  const float* Xb = xm + (size_t)dirb * D_MODELC * LLEN + l0 + lr;
  for (int k = 0; k < D_MODELC; k += 4) {
    v2f a, bb;
    a.x  = Wr[k + 2 * half];
    a.y  = Wr[k + 2 * half + 1];
    bb.x = Xb[(size_t)(k + 2 * half) * LLEN];
    bb.y = Xb[(size_t)(k + 2 * half + 1) * LLEN];
    acc = wmma4(a, bb, acc);
  }
  for (int r = 0; r < 8; ++r) {
    int o = o0 + r + 8 * half;
    xz[((size_t)dirb * 2 * D_INNERC + o) * LLEN + l0 + lr] = acc[r];
  }
}

// ---------- K6: causal depthwise conv1d (K=4) + bias + SiLU ----------
__global__ void k_conv1d(const float* __restrict__ xz, const float* __restrict__ cw,
                         const float* __restrict__ cb, float* __restrict__ xc) {
  int idx = blockIdx.x * blockDim.x + threadIdx.x;
  if (idx >= 16 * D_INNERC * LLEN) return;
  int l = idx & 4095, t = idx >> 12;
  int c = t % D_INNERC, dirb = t / D_INNERC, dir = dirb >> 2;
  const float* src = xz + ((size_t)dirb * 2 * D_INNERC + c) * LLEN;
  const float* w = cw + ((size_t)dir * D_INNERC + c) * D_CONVC;
  float s = cb[dir * D_INNERC + c];
  for (int j = 0; j < D_CONVC; ++j) {
    int lj = l - (D_CONVC - 1) + j;
    if (lj >= 0) s += w[j] * src[lj];
  }
  xc[idx] = silu_(s);
}

// ---------- K7: x-proj GEMM (192 -> 38, padded to 48) ----------
__global__ void k_xproj(const float* __restrict__ xc, const float* __restrict__ xpw,
                        float* __restrict__ dbl) {
  const int lane = threadIdx.x, lr = lane & 15, half = lane >> 4;
  const int l0 = blockIdx.x * 16, o0 = blockIdx.y * 16;
  const int dirb = blockIdx.z, dir = dirb >> 2;
  v8f acc;
  for (int r = 0; r < 8; ++r) acc[r] = 0.f;
  const int orow = o0 + lr;
  const float* Wr = xpw + ((size_t)dir * NDBL + orow) * D_INNERC;
  const float* Xb = xc + (size_t)dirb * D_INNERC * LLEN + l0 + lr;
  for (int k = 0; k < D_INNERC; k += 4) {
    v2f a, bb;
    a.x  = (orow < NDBL) ? Wr[k + 2 * half] : 0.f;
    a.y  = (orow < NDBL) ? Wr[k + 2 * half + 1] : 0.f;
    bb.x = Xb[(size_t)(k + 2 * half) * LLEN];
    bb.y = Xb[(size_t)(k + 2 * half + 1) * LLEN];
    acc = wmma4(a, bb, acc);
  }
  for (int r = 0; r < 8; ++r) {
    int o = o0 + r + 8 * half;
    if (o < NDBL)
      dbl[((size_t)dirb * LLEN + l0 + lr) * NDBL + o] = acc[r];
  }
}

// ---------- K8: selective scan; dt-GEMM+softplus, D-skip and z-gating fused ----------
__global__ void k_scan(const float* __restrict__ dbl, const float* __restrict__ xc,
                       const float* __restrict__ dtw, const float* __restrict__ dtb,
                       const float* __restrict__ Alog, const float* __restrict__ Dp,
                       float* __restrict__ xz) {
  __shared__ float sh[NDBL];
  const int dirb = blockIdx.x, dir = dirb >> 2;
  const int d = threadIdx.x;  // 0..191
  float A[D_STATEC], h[D_STATEC], w6[DT_RANKC];
  for (int n = 0; n < D_STATEC; ++n) {
    A[n] = -__expf(Alog[((size_t)dir * D_INNERC + d) * D_STATEC + n]);
    h[n] = 0.f;
  }
  for (int r = 0; r < DT_RANKC; ++r)
    w6[r] = dtw[((size_t)dir * D_INNERC + d) * DT_RANKC + r];
  const float dtbias = dtb[dir * D_INNERC + d];
  const float Dpd = Dp[dir * D_INNERC + d];
  const float* dblrow = dbl + (size_t)dirb * LLEN * NDBL;
  const float* xrow = xc + ((size_t)dirb * D_INNERC + d) * LLEN;
  const float* zrow = xz + ((size_t)dirb * 2 * D_INNERC + D_INNERC + d) * LLEN;
  float* yrow = xz + ((size_t)dirb * 2 * D_INNERC + d) * LLEN;  // overwrite dead xs half
  for (int l = 0; l < LLEN; ++l) {
    __syncthreads();
    if (d < NDBL) sh[d] = dblrow[(size_t)l * NDBL + d];
    __syncthreads();
    float v = dtbias;
    for (int r = 0; r < DT_RANKC; ++r) v += w6[r] * sh[r];
    float dt = (v > 20.f) ? v : log1pf(__expf(v));
    float x = xrow[l];
    float y = 0.f;
    for (int n = 0; n < D_STATEC; ++n) {
      h[n] = h[n] * __expf(dt * A[n]) + dt * sh[DT_RANKC + n] * x;
      y += h[n] * sh[DT_RANKC + D_STATEC + n];
    }
    y += x * Dpd;
    float z = zrow[l];
    yrow[l] = y * silu_(z);
  }
}

// ---------- K9: out-projection GEMM (192 -> 96) ----------
__global__ void k_outproj(const float* __restrict__ xz, const float* __restrict__ ow,
                          float* __restrict__ Y) {
  const int lane = threadIdx.x, lr = lane & 15, half = lane >> 4;
  const int l0 = blockIdx.x * 16, o0 = blockIdx.y * 16;
  const int dirb = blockIdx.z, dir = dirb >> 2;
  v8f acc;
  for (int r = 0; r < 8; ++r) acc[r] = 0.f;
  const float* Wr = ow + ((size_t)dir * D_MODELC + (o0 + lr)) * D_INNERC;
  const float* Xb = xz + (size_t)dirb * 2 * D_INNERC * LLEN + l0 + lr;
  for (int k = 0; k < D_INNERC; k += 4) {
    v2f a, bb;
    a.x  = Wr[k + 2 * half];
    a.y  = Wr[k + 2 * half + 1];
    bb.x = Xb[(size_t)(k + 2 * half) * LLEN];
    bb.y = Xb[(size_t)(k + 2 * half + 1) * LLEN];
    acc = wmma4(a, bb, acc);
  }
  for (int r = 0; r < 8; ++r) {
    int o = o0 + r + 8 * half;
    Y[((size_t)dirb * LLEN + l0 + lr) * D_MODELC + o] = acc[r];
  }
}

// ---------- K10: layernorm over channels (96) ----------
__global__ void k_lnorm(float* __restrict__ Y, const float* __restrict__ g,
                        const float* __restrict__ be) {
  int row = blockIdx.x * blockDim.x + threadIdx.x;
  if (row >= 16 * LLEN) return;
  float* p = Y + (size_t)row * D_MODELC;
  float s = 0.f;
  for (int c = 0; c < D_MODELC; ++c) s += p[c];
  float mu = s / D_MODELC;
  float vs = 0.f;
  for (int c = 0; c < D_MODELC; ++c) { float d = p[c] - mu; vs += d * d; }
  float rstd = rsqrtf(vs / D_MODELC + EPSV);
  for (int c = 0; c < D_MODELC; ++c) p[c] = (p[c] - mu) * rstd * g[c] + be[c];
}

// ---------- K11: merge 4 directions back to (B,C,H,W) ----------
__global__ void k_combine(const float* __restrict__ Y, float* __restrict__ Ft) {
  int idx = blockIdx.x * blockDim.x + threadIdx.x;
  if (idx >= NBATCH * D_MODELC * HWSZ) return;
  int hw = idx & 4095, bc = idx >> 12;
  int c = bc % D_MODELC, b = bc / D_MODELC;
  int t = ((hw & 63) << 6) | (hw >> 6);
  float s = Y[(((size_t)(0 * 4 + b)) * LLEN + hw) * D_MODELC + c]
          + Y[(((size_t)(1 * 4 + b)) * LLEN + (LLEN - 1 - hw)) * D_MODELC + c]
          + Y[(((size_t)(2 * 4 + b)) * LLEN + t) * D_MODELC + c]
          + Y[(((size_t)(3 * 4 + b)) * LLEN + (LLEN - 1 - t)) * D_MODELC + c];
  Ft[idx] = s;
}

// ---------- K12: final 1x1 conv GEMM + bias + Delta residual ----------
__global__ void k_final(const float* __restrict__ Ft, const float* __restrict__ W,
                        const float* __restrict__ bias, const float* __restrict__ Delta,
                        float* __restrict__ out) {
  const int lane = threadIdx.x, lr = lane & 15, half = lane >> 4;
  const int p0 = blockIdx.x * 16;
  const int b = p0 >> 12, hw0 = p0 & 4095;
  const int o0 = blockIdx.y * 16;
  v8f acc;
  for (int r = 0; r < 8; ++r) acc[r] = bias[o0 + r + 8 * half];
  const float* Wr = W + (size_t)(o0 + lr) * D_MODELC;
  const float* Xb = Ft + (size_t)b * D_MODELC * HWSZ + hw0 + lr;
  for (int k = 0; k < D_MODELC; k += 4) {
    v2f a, bb;
    a.x  = Wr[k + 2 * half];
    a.y  = Wr[k + 2 * half + 1];
    bb.x = Xb[(size_t)(k + 2 * half) * HWSZ];
    bb.y = Xb[(size_t)(k + 2 * half + 1) * HWSZ];
    acc = wmma4(a, bb, acc);
  }
  for (int r = 0; r < 8; ++r) {
    int o = o0 + r + 8 * half;
    size_t oi = ((size_t)b * D_MODELC + o) * HWSZ + hw0 + lr;
    out[oi] = acc[r] + Delta[oi];
  }
}

extern "C" void kernel_launch(void* const* d_in, const int* in_sizes, int n_in,
                              void* d_out, int out_size, void* d_ws, size_t ws_size,
                              hipStream_t stream) {
  (void)in_sizes; (void)n_in; (void)out_size; (void)ws_size;
  const float* Fs    = (const float*)d_in[0];
  const float* HFs   = (const float*)d_in[1];
  const float* Gs    = (const float*)d_in[2];
  const float* Delta = (const float*)d_in[3];
  const float* pfw1  = (const float*)d_in[4];
  const float* pfb1  = (const float*)d_in[5];
  const float* pfdw  = (const float*)d_in[6];
  const float* pfb2  = (const float*)d_in[7];
  const float* phw1  = (const float*)d_in[8];
  const float* phb1  = (const float*)d_in[9];
  const float* phdw  = (const float*)d_in[10];
  const float* phb2  = (const float*)d_in[11];
  const float* gamma = (const float*)d_in[12];
  const float* hfw   = (const float*)d_in[13];
  const float* hfb   = (const float*)d_in[14];
  const float* inw   = (const float*)d_in[15];
  const float* cw    = (const float*)d_in[16];
  const float* cb    = (const float*)d_in[17];
  const float* xpw   = (const float*)d_in[18];
  const float* dtw   = (const float*)d_in[19];
  const float* dtb   = (const float*)d_in[20];
  const float* Alog  = (const float*)d_in[21];
  const float* Dp    = (const float*)d_in[22];
  const float* ow    = (const float*)d_in[23];
  const float* lng   = (const float*)d_in[24];
  const float* lnb   = (const float*)d_in[25];
  const float* opw   = (const float*)d_in[26];
  const float* opb   = (const float*)d_in[27];
  float* out = (float*)d_out;

  float* ws = (float*)d_ws;
  const size_t SZ = (size_t)NBATCH * D_MODELC * HWSZ;      // 1,572,864 floats
  float* t1   = ws;                                        // 2*SZ (dead after K2)
  float* dbl  = ws;                                        // reuses t1 region
  float* Pf   = ws + 2 * SZ;                               // SZ
  float* Phb  = Pf + SZ;                                   // SZ
  float* xm   = Phb + SZ;                                  // 16*96*L (dead after K5)
  float* Ybuf = xm;                                        // reuses xm region
  float* xz   = xm + (size_t)16 * D_MODELC * LLEN;         // 16*384*L
  float* xc   = xz + (size_t)16 * 2 * D_INNERC * LLEN;     // 16*192*L
  float* Ft   = xc + (size_t)16 * D_INNERC * LLEN;         // SZ  (total ~208 MB)

  k_proj_in <<<dim3(1024, 6, 2), 32, 0, stream>>>(Fs, HFs, pfw1, pfb1, phw1, phb1, t1);
  k_dwconv  <<<dim3(6144, 2),   256, 0, stream>>>(t1, pfdw, pfb2, phdw, phb2, Pf, Phb);
  k_instnorm<<<384,             256, 0, stream>>>(Phb, Gs, gamma);
  k_gate    <<<dim3(256, 6, 16), 32, 0, stream>>>(Pf, Phb, hfw, hfb, xm);
  k_inproj  <<<dim3(256, 24, 16),32, 0, stream>>>(xm, inw, xz);
  k_conv1d  <<<49152,           256, 0, stream>>>(xz, cw, cb, xc);
  k_xproj   <<<dim3(256, 3, 16), 32, 0, stream>>>(xc, xpw, dbl);
  k_scan    <<<16,              192, 0, stream>>>(dbl, xc, dtw, dtb, Alog, Dp, xz);
  k_outproj <<<dim3(256, 6, 16), 32, 0, stream>>>(xz, ow, Ybuf);
  k_lnorm   <<<256,             256, 0, stream>>>(Ybuf, lng, lnb);
  k_combine <<<6144,            256, 0, stream>>>(Ybuf, Ft);
  k_final   <<<dim3(1024, 6),    32, 0, stream>>>(Ft, opw, opb, Delta, out);
}